// GCN_79714593014201
// MI455X (gfx1250) — compile-verified
//
#include <hip/hip_runtime.h>
#include <cstdint>

// ---------------------------------------------------------------------------
// GCN 2-layer forward for MI455X (gfx1250, wave32).
// - GEMMs on the fp32 matrix pipe: V_WMMA_F32_16X16X4_F32, 2 row-bands per
//   wave so each B fragment feeds 2 WMMAs.
// - Aggregation via per-call CSR build (count + LDS scan + reorder), then a
//   conflict-free per-node gather-sum with the epilogue (self loop, bias,
//   ReLU, exact-JAX dropout) fused in. No fp32 atomics at all.
// ---------------------------------------------------------------------------

typedef float v2f __attribute__((ext_vector_type(2)));
typedef float v8f __attribute__((ext_vector_type(8)));

#define DEV __device__ __forceinline__

DEV uint32_t rotl32(uint32_t x, uint32_t r) { return (x << r) | (x >> (32u - r)); }

// Exact JAX threefry2x32 for key = jax.random.key(42)  ->  (k0,k1) = (0,42)
DEV uint32_t threefry_42(uint32_t c0, uint32_t c1, bool take_x0) {
  const uint32_t k0 = 0u, k1 = 42u;
  const uint32_t k2 = 0x1BD11BDAu ^ k0 ^ k1;
  const uint32_t ks[3] = {k0, k1, k2};
  uint32_t x0 = c0 + k0, x1 = c1 + k1;
#define TFR(r) { x0 += x1; x1 = rotl32(x1, (r)); x1 ^= x0; }
  TFR(13) TFR(15) TFR(26) TFR(6)   x0 += ks[1]; x1 += ks[2] + 1u;
  TFR(17) TFR(29) TFR(16) TFR(24)  x0 += ks[2]; x1 += ks[0] + 2u;
  TFR(13) TFR(15) TFR(26) TFR(6)   x0 += ks[0]; x1 += ks[1] + 3u;
  TFR(17) TFR(29) TFR(16) TFR(24)  x0 += ks[1]; x1 += ks[2] + 4u;
  TFR(13) TFR(15) TFR(26) TFR(6)   x0 += ks[2]; x1 += ks[0] + 5u;
#undef TFR
  return take_x0 ? x0 : x1;
}

// JAX bernoulli(p=0.5) over flat index i of an array with `total` elements.
DEV bool dropout_keep(uint32_t i, uint32_t total) {
  const uint32_t half = total >> 1;
  uint32_t bits = (i < half) ? threefry_42(i, i + half, true)
                             : threefry_42(i - half, i, false);
  float u = __uint_as_float((bits >> 9) | 0x3F800000u) - 1.0f;
  return u < 0.5f;
}

// ---------------------------------------------------------------------------
// small utilities
// ---------------------------------------------------------------------------
__global__ void izero_kernel(int* __restrict__ p, int n) {
  int i = blockIdx.x * blockDim.x + threadIdx.x;
  if (i < n) p[i] = 0;
}

__global__ void count_kernel(const int* __restrict__ dst, int* __restrict__ cnt, int nE) {
  int e = blockIdx.x * blockDim.x + threadIdx.x;
  if (e < nE) atomicAdd(&cnt[dst[e]], 1);
}

__global__ void dinv_kernel(const int* __restrict__ cnt, float* __restrict__ dinv, int n) {
  int i = blockIdx.x * blockDim.x + threadIdx.x;
  if (i < n) dinv[i] = rsqrtf((float)cnt[i] + 1.0f);
}

// single-block exclusive scan: rowptr[0..n] from cnt[0..n-1]
__global__ __launch_bounds__(1024) void scan_kernel(const int* __restrict__ cnt,
                                                    int* __restrict__ rowptr, int n) {
  __shared__ int sums[1024];
  const int t = threadIdx.x;
  const int chunk = (n + 1023) / 1024;
  const int lo = t * chunk;
  const int hi = min(lo + chunk, n);
  int s = 0;
  for (int i = lo; i < hi; ++i) s += cnt[i];
  sums[t] = s;
  __syncthreads();
  // Hillis-Steele inclusive scan over the 1024 chunk sums
  for (int off = 1; off < 1024; off <<= 1) {
    int v = (t >= off) ? sums[t - off] : 0;
    __syncthreads();
    sums[t] += v;
    __syncthreads();
  }
  int run = (t == 0) ? 0 : sums[t - 1];
  for (int i = lo; i < hi; ++i) { rowptr[i] = run; run += cnt[i]; }
  if (t == 0) rowptr[n] = sums[1023];
}

__global__ void copy_ctr_kernel(const int* __restrict__ rowptr, int* __restrict__ ctr, int n) {
  int i = blockIdx.x * blockDim.x + threadIdx.x;
  if (i < n) ctr[i] = rowptr[i];
}

// bucket edges by dst; precompute per-edge coefficient dinv[src]*dinv[dst]
__global__ void reorder_kernel(const int* __restrict__ src, const int* __restrict__ dst,
                               const float* __restrict__ dinv, int* __restrict__ ctr,
                               int* __restrict__ esrc, float* __restrict__ ecoef, int nE) {
  int e = blockIdx.x * blockDim.x + threadIdx.x;
  if (e >= nE) return;
  int s = src[e], d = dst[e];
  int pos = atomicAdd(&ctr[d], 1);
  esrc[pos] = s;
  ecoef[pos] = dinv[s] * dinv[d];
}

// ---------------------------------------------------------------------------
// H[M x N] = X[M x K] * W[K x N]  with V_WMMA_F32_16X16X4_F32, wave32.
// One wave computes TWO 16-row bands across all N columns so every B fragment
// is reused by two WMMAs.  Band indices are wave-uniform -> EXEC stays all-1.
// ---------------------------------------------------------------------------
template <int NCOL, int KDIM>
__global__ __launch_bounds__(128) void gemm_wmma_f32(
    const float* __restrict__ X, const float* __restrict__ W,
    float* __restrict__ H, int M) {
  const int lane = threadIdx.x & 31;
  const int wave = threadIdx.x >> 5;
  const int nBands = (M + 15) >> 4;
  const int b0 = (blockIdx.x * 4 + wave) * 2;
  if (b0 >= nBands) return;
  const int b1 = min(b0 + 1, nBands - 1);      // clamp (redundant compute, store masked)
  const bool storeB1 = (b0 + 1 < nBands);

  const int r    = lane & 15;
  const int koff = (lane >> 4) << 1;           // 0 (lanes 0-15) or 2 (lanes 16-31)

  const float* __restrict__ xrow0 = X + (size_t)(b0 * 16 + r) * KDIM;
  const float* __restrict__ xrow1 = X + (size_t)(b1 * 16 + r) * KDIM;

  constexpr int NT = NCOL / 16;
  v8f acc0[NT] = {};
  v8f acc1[NT] = {};

  for (int k = 0; k < KDIM; k += 4) {
    const v2f a0 = *(const v2f*)(xrow0 + k + koff);
    const v2f a1 = *(const v2f*)(xrow1 + k + koff);
#pragma unroll
    for (int t = 0; t < NT; ++t) {
      const float* wp = W + (size_t)(k + koff) * NCOL + t * 16 + r;
      v2f b;
      b.x = wp[0];
      b.y = wp[NCOL];
      acc0[t] = __builtin_amdgcn_wmma_f32_16x16x4_f32(
          false, a0, false, b, (short)0, acc0[t], false, false);
      acc1[t] = __builtin_amdgcn_wmma_f32_16x16x4_f32(
          false, a1, false, b, (short)0, acc1[t], false, false);
    }
  }

  const int hiHalf = (lane >> 4) * 8;          // C/D: VGPR v -> M=v (+8 hi lanes)
#pragma unroll
  for (int t = 0; t < NT; ++t) {
#pragma unroll
    for (int v = 0; v < 8; ++v) {
      H[(size_t)(b0 * 16 + hiHalf + v) * NCOL + t * 16 + r] = acc0[t][v];
    }
  }
  if (storeB1) {
#pragma unroll
    for (int t = 0; t < NT; ++t) {
#pragma unroll
      for (int v = 0; v < 8; ++v) {
        H[(size_t)(b1 * 16 + hiHalf + v) * NCOL + t * 16 + r] = acc1[t][v];
      }
    }
  }
}

// ---------------------------------------------------------------------------
// Per-node CSR gather-sum with fused epilogue.
// Block = one node, CH threads (one per channel). Edge (src,coef) pairs are
// staged through LDS in CH-sized tiles; the h[src] row reads are coalesced
// across the block and L2-resident.
//   out = sum_e coef_e * h[src_e]  +  h[node]*dinv^2  +  bias
//   (+ ReLU + dropout for layer 1)
// ---------------------------------------------------------------------------
template <int CH, bool DROPOUT>
__global__ __launch_bounds__(CH) void gather_kernel(
    const float* __restrict__ h, const int* __restrict__ rowptr,
    const int* __restrict__ esrc, const float* __restrict__ ecoef,
    const float* __restrict__ dinv, const float* __restrict__ bias,
    float* __restrict__ out, int n) {
  __shared__ int   s_src[CH];
  __shared__ float s_w[CH];

  const int node = blockIdx.x;
  if (node >= n) return;
  const int c  = threadIdx.x;
  const int lo = rowptr[node];
  const int hi = rowptr[node + 1];

  float acc = 0.0f;
  for (int base = lo; base < hi; base += CH) {
    const int m = min(CH, hi - base);
    if (c < m) {
      s_src[c] = esrc[base + c];
      s_w[c]   = ecoef[base + c];
    }
    __syncthreads();
    for (int j = 0; j < m; ++j) {
      acc = fmaf(h[(size_t)s_src[j] * CH + c], s_w[j], acc);
    }
    __syncthreads();
  }

  const float di = dinv[node];
  float v = acc + h[(size_t)node * CH + c] * (di * di) + bias[c];
  const uint32_t i = (uint32_t)node * CH + c;
  if (DROPOUT) {
    v = fmaxf(v, 0.0f);
    v = dropout_keep(i, (uint32_t)n * CH) ? v * 2.0f : 0.0f;
  }
  out[i] = v;
}

// ---------------------------------------------------------------------------
// launch
// ---------------------------------------------------------------------------
extern "C" void kernel_launch(void* const* d_in, const int* in_sizes, int n_in,
                              void* d_out, int out_size, void* d_ws, size_t ws_size,
                              hipStream_t stream) {
  (void)n_in; (void)out_size; (void)ws_size;
  constexpr int IN_CH = 256, HID = 128, OUT = 64;

  const float* x  = (const float*)d_in[0];
  const int*   ei = (const int*)d_in[1];
  const float* W1 = (const float*)d_in[2];
  const float* b1 = (const float*)d_in[3];
  const float* W2 = (const float*)d_in[4];
  const float* b2 = (const float*)d_in[5];
  float* out = (float*)d_out;

  const int n  = in_sizes[0] / IN_CH;   // 50000 nodes
  const int nE = in_sizes[1] / 2;       // 800000 edges
  const int* src = ei;
  const int* dst = ei + nE;

  // ---- workspace carve-up (256B aligned regions) ----
  char* wp = (char*)d_ws;
  auto carve = [&](size_t bytes) -> void* {
    void* p = (void*)wp;
    wp += (bytes + 255) & ~(size_t)255;
    return p;
  };
  int*   cnt    = (int*)  carve((size_t)n * 4);
  int*   rowptr = (int*)  carve((size_t)(n + 1) * 4);
  int*   ctr    = (int*)  carve((size_t)n * 4);
  int*   esrc   = (int*)  carve((size_t)nE * 4);
  float* ecoef  = (float*)carve((size_t)nE * 4);
  float* dinv   = (float*)carve((size_t)n * 4);
  float* h1     = (float*)carve((size_t)n * HID * 4);
  float* h1d    = (float*)carve((size_t)n * HID * 4);
  float* h2     = h1;                   // reuse: h1 dead after gather1

  const int T = 256;

  // 1) CSR build (re-done every call: deterministic, no cross-call state)
  izero_kernel<<<(n + T - 1) / T, T, 0, stream>>>(cnt, n);
  count_kernel<<<(nE + T - 1) / T, T, 0, stream>>>(dst, cnt, nE);
  scan_kernel<<<1, 1024, 0, stream>>>(cnt, rowptr, n);
  dinv_kernel<<<(n + T - 1) / T, T, 0, stream>>>(cnt, dinv, n);
  copy_ctr_kernel<<<(n + T - 1) / T, T, 0, stream>>>(rowptr, ctr, n);
  reorder_kernel<<<(nE + T - 1) / T, T, 0, stream>>>(src, dst, dinv, ctr, esrc, ecoef, nE);

  // 2) h1 = X * W1   (WMMA f32, 2 bands/wave)
  {
    const int nBands = (n + 15) / 16;
    const int slots  = (nBands + 1) / 2;
    gemm_wmma_f32<HID, IN_CH><<<(slots + 3) / 4, 128, 0, stream>>>(x, W1, h1, n);
  }

  // 3) h1d = dropout(relu(gather(h1) + self + b1))
  gather_kernel<HID, true><<<n, HID, 0, stream>>>(h1, rowptr, esrc, ecoef, dinv, b1, h1d, n);

  // 4) h2 = h1d * W2   (WMMA f32)
  {
    const int nBands = (n + 15) / 16;
    const int slots  = (nBands + 1) / 2;
    gemm_wmma_f32<OUT, HID><<<(slots + 3) / 4, 128, 0, stream>>>(h1d, W2, h2, n);
  }

  // 5) out = gather(h2) + self + b2
  gather_kernel<OUT, false><<<n, OUT, 0, stream>>>(h2, rowptr, esrc, ecoef, dinv, b2, out, n);
}